// FastRCNNInference_12919261626791
// MI455X (gfx1250) — compile-verified
//
#include <hip/hip_runtime.h>
#include <stdint.h>
#include <math.h>

// ---- problem constants (match reference) ----
#define BB   16
#define PP   2000
#define CC   81
#define NCC  80          // C-1
#define DD   100
#define TT   100
#define PPAD 2048
#define NEGV (-1.0e10f)
#define BBOX_CLIP_F 4.135166556742356f   // log(1000/16)

// ---------------------------------------------------------------------------
// CDNA5 async global->LDS path (guarded; falls back to plain loads if the
// builtins are absent on this toolchain). Round-1 diagnostics confirmed the
// builtin exists here with (global int4*, LDS int4*, imm, imm) signature.
// ---------------------------------------------------------------------------
#ifdef __has_builtin
# if __has_builtin(__builtin_amdgcn_global_load_async_to_lds_b128)
#  define HAVE_ASYNC_LDS 1
# endif
# if __has_builtin(__builtin_amdgcn_s_wait_asynccnt)
#  define HAVE_WAIT_ASYNC_BUILTIN 1
# endif
#endif
#ifndef HAVE_ASYNC_LDS
# define HAVE_ASYNC_LDS 0
#endif

#if HAVE_ASYNC_LDS
typedef int v4i_t __attribute__((ext_vector_type(4)));
typedef __attribute__((address_space(1))) v4i_t gv4i_t;   // global int4
typedef __attribute__((address_space(3))) v4i_t lv4i_t;   // LDS int4
__device__ __forceinline__ void async_copy_b128(const void* g, void* l) {
  // Generic->AS pointers via integer casts: global flat addr == global addr;
  // LDS flat addr low 32 bits == LDS byte offset (ISA 10.2 aperture rules).
  __builtin_amdgcn_global_load_async_to_lds_b128(
      (gv4i_t*)(uintptr_t)g,
      (lv4i_t*)(uint32_t)(uintptr_t)l,
      0, 0);
}
__device__ __forceinline__ void wait_async0() {
#ifdef HAVE_WAIT_ASYNC_BUILTIN
  __builtin_amdgcn_s_wait_asynccnt(0);
#else
  asm volatile("s_wait_asynccnt 0" ::: "memory");
#endif
}
#endif

static __device__ __forceinline__ void argmax_combine(float& bv, int& bi, float ov, int oi) {
  // prefer greater value; on exact tie prefer LOWER index (matches jnp.argmax)
  if (ov > bv || (ov == bv && oi < bi)) { bv = ov; bi = oi; }
}

// ---------------------------------------------------------------------------
// K1: softmax + box decode + clip, transposed to [b][class][p] layout.
// One thread per proposal row (N = B*P = 32000).
// ---------------------------------------------------------------------------
__global__ void __launch_bounds__(256) k_decode(
    const float* __restrict__ logits, const float* __restrict__ reg,
    const float* __restrict__ boxes, float* __restrict__ sc_ws,
    float* __restrict__ box_ws) {
  int row = blockIdx.x * 256 + threadIdx.x;
  if (row >= BB * PP) return;
  const float* lrow = logits + (size_t)row * CC;
  float mx = lrow[0];
  for (int c = 1; c < CC; ++c) mx = fmaxf(mx, lrow[c]);
  float sum = 0.0f;
  for (int c = 0; c < CC; ++c) sum += expf(lrow[c] - mx);

  float4 bx = ((const float4*)boxes)[row];
  float w  = bx.z - bx.x + 1.0f;
  float h  = bx.w - bx.y + 1.0f;
  float cx = bx.x + 0.5f * w;
  float cy = bx.y + 0.5f * h;

  int b = row / PP;
  int p = row - b * PP;
  const float4* regr = (const float4*)(reg + (size_t)row * 4 * CC);

  for (int c = 1; c < CC; ++c) {
    float s  = expf(lrow[c] - mx) / sum;
    float so = (s > 0.05f) ? s : NEGV;            // threshold applied up front
    float4 d = regr[c];
    float dx = d.x / 10.0f;
    float dy = d.y / 10.0f;
    float dw = fminf(d.z / 5.0f, BBOX_CLIP_F);
    float dh = fminf(d.w / 5.0f, BBOX_CLIP_F);
    float pcx = dx * w + cx;
    float pcy = dy * h + cy;
    float pw  = expf(dw) * w;
    float ph  = expf(dh) * h;
    float o0 = fminf(fmaxf(pcx - 0.5f * pw, 0.0f), 1332.0f);
    float o1 = fminf(fmaxf(pcy - 0.5f * ph, 0.0f), 799.0f);
    float o2 = fminf(fmaxf(pcx + 0.5f * pw - 1.0f, 0.0f), 1332.0f);
    float o3 = fminf(fmaxf(pcy + 0.5f * ph - 1.0f, 0.0f), 799.0f);
    size_t oi = ((size_t)(b * NCC + (c - 1))) * PP + p;   // [b][cls][p]
    sc_ws[oi] = so;
    ((float4*)box_ws)[oi] = make_float4(o0, o1, o2, o3);
  }
}

// ---------------------------------------------------------------------------
// K2: NMS. One 256-thread block (8 wave32) per (batch, class) problem.
// 1) Async-stage scores+boxes into LDS (ASYNCcnt + s_wait_asynccnt).
// 2) Stable compaction of valid (score>thresh) entries -> Nv entries; the
//    100-step argmax/suppress loop then runs over Nv instead of 2048, which
//    removes ~80-90% of the dominant IoU VALU work. Stability preserves the
//    jnp.argmax lower-index tie-break ordering.
// ---------------------------------------------------------------------------
__global__ void __launch_bounds__(256) k_nms(
    const float* __restrict__ sc_ws, const float* __restrict__ box_ws,
    float* __restrict__ cand_s, float* __restrict__ cand_b) {
  __shared__ float4 lbox[PPAD];        // boxes (AoS), 32KB
  __shared__ float4 ls4[PPAD / 4];     // scores, float4 for 16B-aligned async
  __shared__ float  larea[PPAD];
  __shared__ int    cnts[256];
  __shared__ float  r_v[8];
  __shared__ int    r_i[8];
  __shared__ float  bc_v;
  __shared__ int    bc_i;
  float* ls = (float*)ls4;

  int bc  = blockIdx.x;                // bc = b*80 + cls
  int tid = threadIdx.x;
  const float*  sbase = sc_ws + (size_t)bc * PP;
  const float4* bbase = (const float4*)box_ws + (size_t)bc * PP;

#if HAVE_ASYNC_LDS
  for (int j = tid; j < PP / 4; j += 256)          // 500 x b128 score chunks
    async_copy_b128(sbase + j * 4, &ls[j * 4]);
  for (int j = tid; j < PP; j += 256)              // 2000 x b128 box chunks
    async_copy_b128(bbase + j, &lbox[j]);
  wait_async0();
  __syncthreads();
#else
  for (int p = tid; p < PP; p += 256) {
    ls[p]   = sbase[p];
    lbox[p] = bbase[p];
  }
  __syncthreads();
#endif

  // pad original tail so compaction reads are well-defined
  for (int p = PP + tid; p < PPAD; p += 256) {
    ls[p]   = NEGV;
    lbox[p] = make_float4(0.0f, 0.0f, 0.0f, 0.0f);
  }
  __syncthreads();

  // ---- stable compaction (thread tid owns contiguous [tid*8, tid*8+8)) ----
  float  rv[8];
  float4 rb[8];
  int base = tid * 8, cnt = 0;
#pragma unroll
  for (int j = 0; j < 8; ++j) {
    rv[j] = ls[base + j];
    rb[j] = lbox[base + j];
    cnt += (rv[j] > NEGV * 0.5f) ? 1 : 0;
  }
  __syncthreads();                   // all reads done before in-place writes

  cnts[tid] = cnt;
  __syncthreads();
  for (int off = 1; off < 256; off <<= 1) {        // Hillis-Steele incl. scan
    int v = cnts[tid];
    int o = (tid >= off) ? cnts[tid - off] : 0;
    __syncthreads();
    cnts[tid] = v + o;
    __syncthreads();
  }
  int Nv  = cnts[255];
  int pos = cnts[tid] - cnt;                        // exclusive prefix
#pragma unroll
  for (int j = 0; j < 8; ++j) {
    if (rv[j] > NEGV * 0.5f) {
      ls[pos]   = rv[j];
      lbox[pos] = rb[j];
      ++pos;
    }
  }
  __syncthreads();

  // clear tail beyond compacted region; compute areas of compacted boxes
  for (int p = Nv + tid; p < PPAD; p += 256) ls[p] = NEGV;
  for (int p = tid; p < PPAD; p += 256) {
    if (p < Nv) {
      float4 q = lbox[p];
      larea[p] = (q.z - q.x + 1.0f) * (q.w - q.y + 1.0f);
    } else {
      larea[p] = 0.0f;
    }
  }
  __syncthreads();

  if (Nv == 0) {                      // nothing above threshold: all invalid
    for (int tt = tid; tt < TT; tt += 256) {
      cand_s[(size_t)bc * TT + tt] = NEGV;
      ((float4*)cand_b)[(size_t)bc * TT + tt] =
          make_float4(0.0f, 0.0f, 0.0f, 0.0f);
    }
    return;
  }

  int NvPad = (Nv + 255) & ~255;      // >= 256, multiple of 256
  int wav = tid >> 5, lane = tid & 31;

  for (int t = 0; t < TT; ++t) {
    // ---- block argmax over compacted entries, first-index tie-break ----
    float bv = -3.0e38f;
    int   bi = PPAD;
    for (int p = tid; p < NvPad; p += 256) {        // ascending p per thread
      float v = ls[p];
      if (v > bv) { bv = v; bi = p; }               // strict > keeps lowest
    }
#pragma unroll
    for (int off = 16; off > 0; off >>= 1) {
      float ov = __shfl_xor(bv, off, 32);
      int   oi = __shfl_xor(bi, off, 32);
      argmax_combine(bv, bi, ov, oi);
    }
    if (lane == 0) { r_v[wav] = bv; r_i[wav] = bi; }
    __syncthreads();
    if (tid == 0) {
      float v = r_v[0]; int i = r_i[0];
      for (int q = 1; q < 8; ++q) argmax_combine(v, i, r_v[q], r_i[q]);
      bc_v = v; bc_i = i;
    }
    __syncthreads();

    float si = bc_v;
    int   ii = bc_i;
    float4 bxx = lbox[ii];

    if (si <= NEGV * 0.5f) {
      // Exhausted: reference keeps recording (NEG, frozen argmax) — those
      // entries are masked to zero downstream; any deterministic box is fine.
      for (int tt = t + tid; tt < TT; tt += 256) {
        cand_s[(size_t)bc * TT + tt] = NEGV;
        ((float4*)cand_b)[(size_t)bc * TT + tt] = bxx;
      }
      break;                           // uniform across block
    }

    float ai = larea[ii];
    if (tid == 0) {
      cand_s[(size_t)bc * TT + t] = si;
      ((float4*)cand_b)[(size_t)bc * TT + t] = bxx;
    }

    // ---- IoU suppression over compacted entries only ----
    for (int p = tid; p < NvPad; p += 256) {
      float4 q = lbox[p];
      float iw = fmaxf(fminf(q.z, bxx.z) - fmaxf(q.x, bxx.x) + 1.0f, 0.0f);
      float ih = fmaxf(fminf(q.w, bxx.w) - fmaxf(q.y, bxx.y) + 1.0f, 0.0f);
      float inter = iw * ih;
      float iou = inter / (larea[p] + ai - inter);
      if (iou > 0.5f || p == ii) ls[p] = NEGV;
    }
    __syncthreads();
  }
}

// ---------------------------------------------------------------------------
// K3: per-batch top-100 of 8000 (iterative selection == stable top_k order),
// final masking + label reconstruction. One block per batch.
// ---------------------------------------------------------------------------
__global__ void __launch_bounds__(256) k_topk(
    const float* __restrict__ cand_s, const float* __restrict__ cand_b,
    float* __restrict__ out) {
  __shared__ float ts[8192];
  __shared__ float r_v[8];
  __shared__ int   r_i[8];
  int b = blockIdx.x, tid = threadIdx.x;
  const float* sb = cand_s + (size_t)b * (NCC * TT);
  for (int j = tid; j < 8192; j += 256)
    ts[j] = (j < NCC * TT) ? sb[j] : -3.4e38f;
  __syncthreads();

  float* out_b = out;                        // [B][D][4]
  float* out_s = out + BB * DD * 4;          // [B][D]
  float* out_l = out + BB * DD * 4 + BB * DD;// [B][D] (labels as floats)

  int wav = tid >> 5, lane = tid & 31;
  for (int t = 0; t < DD; ++t) {
    float bv = ts[tid];
    int   bi = tid;
#pragma unroll
    for (int k = 1; k < 32; ++k) {
      int   p = tid + (k << 8);
      float v = ts[p];
      if (v > bv) { bv = v; bi = p; }
    }
#pragma unroll
    for (int off = 16; off > 0; off >>= 1) {
      float ov = __shfl_xor(bv, off, 32);
      int   oi = __shfl_xor(bi, off, 32);
      argmax_combine(bv, bi, ov, oi);
    }
    if (lane == 0) { r_v[wav] = bv; r_i[wav] = bi; }
    __syncthreads();
    if (tid == 0) {
      float v = r_v[0]; int i = r_i[0];
      for (int q = 1; q < 8; ++q) argmax_combine(v, i, r_v[q], r_i[q]);
      bool valid = v > NEGV * 0.5f;
      out_s[b * DD + t] = valid ? v : 0.0f;
      float4 bxx = ((const float4*)cand_b)[(size_t)b * (NCC * TT) + i];
      ((float4*)out_b)[b * DD + t] =
          valid ? bxx : make_float4(0.0f, 0.0f, 0.0f, 0.0f);
      out_l[b * DD + t] = valid ? (float)(i / TT + 1) : 0.0f;
      ts[i] = -3.4e38f;                 // remove selected entry
    }
    __syncthreads();
  }
}

// ---------------------------------------------------------------------------
extern "C" void kernel_launch(void* const* d_in, const int* in_sizes, int n_in,
                              void* d_out, int out_size, void* d_ws, size_t ws_size,
                              hipStream_t stream) {
  (void)in_sizes; (void)n_in; (void)out_size; (void)ws_size;
  const float* logits = (const float*)d_in[0];
  const float* reg    = (const float*)d_in[1];
  const float* boxes  = (const float*)d_in[2];

  float* ws = (float*)d_ws;
  const size_t SC_N  = (size_t)BB * NCC * PP;      //  2,560,000 floats
  const size_t BOX_N = SC_N * 4;                   // 10,240,000 floats
  const size_t CS_N  = (size_t)BB * NCC * TT;      //    128,000 floats
  float* sc_ws  = ws;                 // thresholded scores [b][cls][p]
  float* box_ws = ws + SC_N;          // decoded clipped boxes [b][cls][p][4]
  float* cand_s = box_ws + BOX_N;     // NMS-selected scores [b][cls*100+t]
  float* cand_b = cand_s + CS_N;      // NMS-selected boxes  [b][cls*100+t][4]

  k_decode<<<(BB * PP + 255) / 256, 256, 0, stream>>>(logits, reg, boxes,
                                                      sc_ws, box_ws);
  k_nms<<<BB * NCC, 256, 0, stream>>>(sc_ws, box_ws, cand_s, cand_b);
  k_topk<<<BB, 256, 0, stream>>>(cand_s, cand_b, (float*)d_out);
}